// Recon_Loss_37452114821777
// MI455X (gfx1250) — compile-verified
//
#include <hip/hip_runtime.h>

typedef __attribute__((ext_vector_type(2))) float v2f;
typedef __attribute__((ext_vector_type(8))) float v8f;

#define N1 256
#define N2 256
#define N3 512
#define R1_ 64
#define R_  32

// workspace layout (floats):
//  [0] term1, [1] term2 accum, [2] term3 accum
//  A  at WS_A  (N1 x 32)
//  B  at WS_B  (N2 x 32)
//  Ct at WS_CT (32 x N3)  -- C transposed for coalesced WMMA B-operand loads
#define WS_A  16
#define WS_B  (16 + N1 * R_)
#define WS_CT (16 + N1 * R_ + N2 * R_)

// ---------------------------------------------------------------------------
// Kernel 1: factor GEMMs + Gram-trick term1, single block of 256 threads.
// ---------------------------------------------------------------------------
__global__ void __launch_bounds__(256)
prep_kernel(const float* __restrict__ A0, const float* __restrict__ B0,
            const float* __restrict__ C0, const float* __restrict__ A_S,
            const float* __restrict__ B_S, const float* __restrict__ C_S,
            float* __restrict__ ws) {
    const int t = threadIdx.x;
    float* A  = ws + WS_A;
    float* B  = ws + WS_B;
    float* Ct = ws + WS_CT;

    // A = A0 @ A_S : row t
    for (int r = 0; r < R_; ++r) {
        float s = 0.f;
        for (int q = 0; q < R1_; ++q) s = fmaf(A0[t * R1_ + q], A_S[q * R_ + r], s);
        A[t * R_ + r] = s;
    }
    // B = B0 @ B_S : row t
    for (int r = 0; r < R_; ++r) {
        float s = 0.f;
        for (int q = 0; q < R1_; ++q) s = fmaf(B0[t * R1_ + q], B_S[q * R_ + r], s);
        B[t * R_ + r] = s;
    }
    // Ct[r,k] = (C0 @ C_S)[k,r] : rows k = t, t+256
    for (int kk = 0; kk < 2; ++kk) {
        const int k = t + kk * 256;
        for (int r = 0; r < R_; ++r) {
            float s = 0.f;
            for (int q = 0; q < R1_; ++q) s = fmaf(C0[k * R1_ + q], C_S[q * R_ + r], s);
            Ct[r * N3 + k] = s;
        }
    }
    __syncthreads();

    // term1 = sum_{r,s} (A'A)_{rs} (B'B)_{rs} (C'C)_{rs}; 4 (r,s) pairs per thread
    float part = 0.f;
    for (int e = 0; e < 4; ++e) {
        const int idx = t * 4 + e;
        const int r = idx >> 5, s = idx & 31;
        float gA = 0.f, gB = 0.f, gC = 0.f;
        for (int i2 = 0; i2 < N1; ++i2) gA = fmaf(A[i2 * R_ + r], A[i2 * R_ + s], gA);
        for (int j2 = 0; j2 < N2; ++j2) gB = fmaf(B[j2 * R_ + r], B[j2 * R_ + s], gB);
        for (int k2 = 0; k2 < N3; ++k2) gC = fmaf(Ct[r * N3 + k2], Ct[s * N3 + k2], gC);
        part += gA * gB * gC;
    }
    __shared__ float red[256];
    red[t] = part;
    __syncthreads();
    for (int off = 128; off > 0; off >>= 1) {
        if (t < off) red[t] += red[t + off];
        __syncthreads();
    }
    if (t == 0) {
        ws[0] = red[0];   // term1
        ws[1] = 0.f;      // term2 accumulator
        ws[2] = 0.f;      // term3 accumulator
    }
}

// ---------------------------------------------------------------------------
// Kernel 2: single fused pass over X. Per wave: D = W(16x32) @ Ct(32x16) via
// 8 chained V_WMMA_F32_16X16X4_F32, then s2 += X.D, s3 += X.X on the tile.
// 4096 blocks x 8 waves; X (134 MB) is read exactly once -> HBM-bound.
// ---------------------------------------------------------------------------
__global__ void __launch_bounds__(256)
recon_kernel(const float* __restrict__ X, float* __restrict__ ws) {
    const float* A  = ws + WS_A;
    const float* B  = ws + WS_B;
    const float* Ct = ws + WS_CT;

    const int lane = threadIdx.x & 31;
    const int wave = threadIdx.x >> 5;
    const int hi   = lane >> 4;     // half-wave select (K split for A/B operands)
    const int n    = lane & 15;

    const int i  = blockIdx.x >> 4;         // 256 i values
    const int j0 = (blockIdx.x & 15) << 4;  // 16 j-tiles per i
    const int j  = j0 + n;

    // A-operand: W[j,r] = A[i,r]*B[j,r]; 16x4 fp32 layout:
    // v0 = K=(base+2*hi), v1 = K=(base+1+2*hi), M = lane&15
    v2f aop[8];
#pragma unroll
    for (int kk = 0; kk < 8; ++kk) {
        const int r0 = kk * 4 + 2 * hi;
        aop[kk].x = A[i * R_ + r0]     * B[j * R_ + r0];
        aop[kk].y = A[i * R_ + r0 + 1] * B[j * R_ + r0 + 1];
    }

    float s2 = 0.f, s3 = 0.f;
#pragma unroll
    for (int kt = 0; kt < 4; ++kt) {
        const int k0 = (wave * 4 + kt) << 4;   // 8 waves x 4 k-tiles = full N3
        v8f acc = {};
#pragma unroll
        for (int kk = 0; kk < 8; ++kk) {
            const int r0 = kk * 4 + 2 * hi;
            v2f bop;
            bop.x = Ct[r0 * N3 + k0 + n];        // B-operand: Ct rows, N = lane&15
            bop.y = Ct[(r0 + 1) * N3 + k0 + n];
            acc = __builtin_amdgcn_wmma_f32_16x16x4_f32(
                false, aop[kk], false, bop, (short)0, acc, false, false);
        }
        // X tile in the C/D layout: M = v + 8*hi, N = lane&15
        const float* Xbase =
            X + ((size_t)i * N2 + (size_t)(j0 + 8 * hi)) * N3 + (size_t)(k0 + n);
#pragma unroll
        for (int v = 0; v < 8; ++v) {
            const float x = Xbase[(size_t)v * N3];
            s2 = fmaf(x, acc[v], s2);
            s3 = fmaf(x, x, s3);
        }
    }

    // wave32 reduction, then one atomic pair per wave
    for (int off = 16; off > 0; off >>= 1) {
        s2 += __shfl_xor(s2, off, 32);
        s3 += __shfl_xor(s3, off, 32);
    }
    if (lane == 0) {
        atomicAdd(&ws[1], s2);
        atomicAdd(&ws[2], s3);
    }
}

// ---------------------------------------------------------------------------
// Kernel 3: loss = (term1 - 2*term2 + term3) / (N1*N2*N3)
// ---------------------------------------------------------------------------
__global__ void finalize_kernel(const float* __restrict__ ws, float* __restrict__ out) {
    out[0] = (ws[0] - 2.0f * ws[1] + ws[2]) *
             (1.0f / ((float)N1 * (float)N2 * (float)N3));
}

extern "C" void kernel_launch(void* const* d_in, const int* in_sizes, int n_in,
                              void* d_out, int out_size, void* d_ws, size_t ws_size,
                              hipStream_t stream) {
    const float* X   = (const float*)d_in[0];
    const float* A0  = (const float*)d_in[1];
    const float* B0  = (const float*)d_in[2];
    const float* C0  = (const float*)d_in[3];
    const float* A_S = (const float*)d_in[4];
    const float* B_S = (const float*)d_in[5];
    const float* C_S = (const float*)d_in[6];
    float* ws  = (float*)d_ws;
    float* out = (float*)d_out;

    prep_kernel<<<1, 256, 0, stream>>>(A0, B0, C0, A_S, B_S, C_S, ws);
    recon_kernel<<<(N1 * N2) / 16, 256, 0, stream>>>(X, ws);
    finalize_kernel<<<1, 1, 0, stream>>>(ws, out);
}